// G2CL_GCNEncoder_80083960201232
// MI455X (gfx1250) — compile-verified
//
#include <hip/hip_runtime.h>
#include <hip/hip_bf16.h>
#include <math.h>

typedef __attribute__((ext_vector_type(2))) float v2f;
typedef __attribute__((ext_vector_type(8))) float v8f;

#define IN_DIM 256
#define HID    128

// ---------------- degree / norm ----------------
__global__ void k_deg_init(float* __restrict__ deg, int N) {
  int i = blockIdx.x * blockDim.x + threadIdx.x;
  if (i < N) deg[i] = 1.0f;  // self loop contributes 1
}

__global__ void k_deg_edges(const long long* __restrict__ col,
                            float* __restrict__ deg, int E) {
  int e = blockIdx.x * blockDim.x + threadIdx.x;
  if (e < E) atomicAdd(&deg[(int)col[e]], 1.0f);
}

__global__ void k_dinv(const float* __restrict__ deg, float* __restrict__ dinv, int N) {
  int i = blockIdx.x * blockDim.x + threadIdx.x;
  if (i < N) {
    float d = deg[i];
    dinv[i] = (d > 0.0f) ? rsqrtf(d) : 0.0f;
  }
}

// ---------------- GEMM 1: t0 = X[M,256] @ W0[256,128]  (f32 WMMA) ----------------
// Block = 256 threads = 8 waves; wave w owns N-tile [16w,16w+16); block owns 16 rows.
__global__ __launch_bounds__(256) void k_gemm_x(const float* __restrict__ X,
                                                const float* __restrict__ W,
                                                float* __restrict__ O, int M) {
  const int K = IN_DIM, N = HID;
  const int lane = threadIdx.x & 31;
  const int wave = threadIdx.x >> 5;
  const int m0 = blockIdx.x * 16;
  const int n0 = wave * 16;
  const int lm = lane & 15;   // A: row in tile / B,C,D: col in tile
  const int hi = lane >> 4;   // K-pair select for A/B; +8 row offset for C/D

  int ma = m0 + lm; if (ma >= M) ma = M - 1;  // clamp (stores are guarded)
  const float* aBase = X + (size_t)ma * K + 2 * hi;
  const float* bBase = W + (size_t)(2 * hi) * N + n0 + lm;

  v8f acc = {};
#pragma unroll 8
  for (int k0 = 0; k0 < K; k0 += 4) {
    v2f a = *(const v2f*)(aBase + k0);              // X[m, k0+2hi .. +1]
    const float* bp = bBase + (size_t)k0 * N;       // W[k0+2hi, n], W[k0+2hi+1, n]
    v2f b; b[0] = bp[0]; b[1] = bp[N];
    acc = __builtin_amdgcn_wmma_f32_16x16x4_f32(false, a, false, b,
                                                (short)0, acc, false, false);
  }

  const int n = n0 + lm;
  float* oBase = O + (size_t)(m0 + hi * 8) * N + n;
  if (m0 + 16 <= M) {           // uniform branch: full tile (always true for M%16==0)
#pragma unroll
    for (int g = 0; g < 8; ++g) oBase[(size_t)g * N] = acc[g];
  } else {
#pragma unroll
    for (int g = 0; g < 8; ++g) {
      if (m0 + hi * 8 + g < M) oBase[(size_t)g * N] = acc[g];
    }
  }
}

// ---------------- aggregation: out = A_norm * in  (two kernels) ----------------
// self-loop term: out[i,:] = dinv[i]^2 * in[i,:]
__global__ void k_selfloop(const float* __restrict__ in, const float* __restrict__ dinv,
                           float* __restrict__ out, int N) {
  int t = blockIdx.x * blockDim.x + threadIdx.x;
  int i = t >> 7, f = t & 127;
  if (i < N) {
    float d = dinv[i];
    out[(size_t)i * HID + f] = d * d * in[(size_t)i * HID + f];
  }
}

// edge term: out[col,:] += dinv[row]*dinv[col] * in[row,:]   (wave per edge)
__global__ void k_scatter(const float* __restrict__ in, float* __restrict__ out,
                          const float* __restrict__ dinv,
                          const long long* __restrict__ row,
                          const long long* __restrict__ col, int E) {
  int t = blockIdx.x * blockDim.x + threadIdx.x;
  int e = t >> 5, lane = t & 31;
  if (e >= E) return;
  int r = (int)row[e], c = (int)col[e];
  float nrm = dinv[r] * dinv[c];
  float4 v = ((const float4*)(in + (size_t)r * HID))[lane];  // 128 floats / 32 lanes
  float* dst = out + (size_t)c * HID + lane * 4;
  atomicAdd(dst + 0, nrm * v.x);
  atomicAdd(dst + 1, nrm * v.y);
  atomicAdd(dst + 2, nrm * v.z);
  atomicAdd(dst + 3, nrm * v.w);
}

__global__ void k_bias_relu(const float* __restrict__ in, const float* __restrict__ b,
                            float* __restrict__ out, int N) {
  int t = blockIdx.x * blockDim.x + threadIdx.x;
  int i = t >> 7, f = t & 127;
  if (i < N) out[t] = fmaxf(in[t] + b[f], 0.0f);
}

// ---------------- GEMM 2 (dual head): mu = g@Wmu+bmu ; sigma = softplus(g@Wsig+bsig)+1e-7
__global__ __launch_bounds__(256) void k_gemm_out(const float* __restrict__ G,
                                                  const float* __restrict__ Wmu,
                                                  const float* __restrict__ bmu,
                                                  const float* __restrict__ Wsig,
                                                  const float* __restrict__ bsig,
                                                  float* __restrict__ out, int M) {
  const int K = HID, N = HID;
  const int lane = threadIdx.x & 31;
  const int wave = threadIdx.x >> 5;
  const int m0 = blockIdx.x * 16;
  const int n0 = wave * 16;
  const int lm = lane & 15;
  const int hi = lane >> 4;

  int ma = m0 + lm; if (ma >= M) ma = M - 1;
  const float* aBase  = G    + (size_t)ma * K + 2 * hi;
  const float* bmBase = Wmu  + (size_t)(2 * hi) * N + n0 + lm;
  const float* bsBase = Wsig + (size_t)(2 * hi) * N + n0 + lm;

  v8f accm = {};
  v8f accs = {};
#pragma unroll 8
  for (int k0 = 0; k0 < K; k0 += 4) {
    v2f a = *(const v2f*)(aBase + k0);
    const float* bmp = bmBase + (size_t)k0 * N;
    const float* bsp = bsBase + (size_t)k0 * N;
    v2f bm; bm[0] = bmp[0]; bm[1] = bmp[N];
    v2f bs; bs[0] = bsp[0]; bs[1] = bsp[N];
    accm = __builtin_amdgcn_wmma_f32_16x16x4_f32(false, a, false, bm,
                                                 (short)0, accm, false, false);
    accs = __builtin_amdgcn_wmma_f32_16x16x4_f32(false, a, false, bs,
                                                 (short)0, accs, false, false);
  }

  const int n = n0 + lm;
  const float bmu_n = bmu[n];
  const float bsg_n = bsig[n];
  float* muBase = out + (size_t)(m0 + hi * 8) * N + n;
  float* sgBase = muBase + (size_t)M * N;

  if (m0 + 16 <= M) {           // uniform branch: full tile
#pragma unroll
    for (int g = 0; g < 8; ++g) {
      float muv = accm[g] + bmu_n;
      float sv  = accs[g] + bsg_n;
      muBase[(size_t)g * N] = muv;
      float sp = fmaxf(sv, 0.0f) + log1pf(expf(-fabsf(sv)));  // stable softplus
      sgBase[(size_t)g * N] = sp + 1e-7f;
    }
  } else {
#pragma unroll
    for (int g = 0; g < 8; ++g) {
      if (m0 + hi * 8 + g < M) {
        float muv = accm[g] + bmu_n;
        float sv  = accs[g] + bsg_n;
        muBase[(size_t)g * N] = muv;
        float sp = fmaxf(sv, 0.0f) + log1pf(expf(-fabsf(sv)));
        sgBase[(size_t)g * N] = sp + 1e-7f;
      }
    }
  }
}

extern "C" void kernel_launch(void* const* d_in, const int* in_sizes, int n_in,
                              void* d_out, int out_size, void* d_ws, size_t ws_size,
                              hipStream_t stream) {
  const float*     x    = (const float*)d_in[0];
  const long long* eidx = (const long long*)d_in[1];   // int64 [2, E] row-major
  const float*     W0   = (const float*)d_in[2];
  const float*     b0   = (const float*)d_in[3];
  const float*     Wmu  = (const float*)d_in[4];
  const float*     bmu  = (const float*)d_in[5];
  const float*     Wsig = (const float*)d_in[6];
  const float*     bsig = (const float*)d_in[7];
  float*           out  = (float*)d_out;

  const int N = in_sizes[0] / IN_DIM;   // 100000 nodes
  const int E = in_sizes[1] / 2;        // 1600000 edges
  const long long* row = eidx;          // sources
  const long long* col = eidx + E;      // targets

  // workspace layout (floats): deg[N] | dinv[N] | bufA[N*128] | bufB[N*128]
  float* ws   = (float*)d_ws;
  float* deg  = ws;
  float* dinv = ws + (size_t)N;
  float* bufA = ws + (size_t)2 * N;                   // t0, later h
  float* bufB = bufA + (size_t)N * HID;               // agg0, later g

  const int T = 256;
  // 1) normalization coefficients
  k_deg_init <<<(N + T - 1) / T, T, 0, stream>>>(deg, N);
  k_deg_edges<<<(E + T - 1) / T, T, 0, stream>>>(col, deg, E);
  k_dinv     <<<(N + T - 1) / T, T, 0, stream>>>(deg, dinv, N);

  // 2) t0 = x @ W0   (WMMA f32)
  k_gemm_x<<<(N + 15) / 16, T, 0, stream>>>(x, W0, bufA, N);

  // 3) agg0 = A_norm * t0
  k_selfloop<<<((size_t)N * HID + T - 1) / T, T, 0, stream>>>(bufA, dinv, bufB, N);
  k_scatter <<<((size_t)E * 32 + T - 1) / T, T, 0, stream>>>(bufA, bufB, dinv, row, col, E);

  // 4) h = relu(agg0 + b0)
  k_bias_relu<<<((size_t)N * HID + T - 1) / T, T, 0, stream>>>(bufB, b0, bufA, N);

  // 5) g = A_norm * h   (shared by both heads: A(hW) == (Ah)W)
  k_selfloop<<<((size_t)N * HID + T - 1) / T, T, 0, stream>>>(bufA, dinv, bufB, N);
  k_scatter <<<((size_t)E * 32 + T - 1) / T, T, 0, stream>>>(bufA, bufB, dinv, row, col, E);

  // 6) mu / sigma heads, fused epilogue -> d_out
  k_gemm_out<<<(N + 15) / 16, T, 0, stream>>>(bufB, Wmu, bmu, Wsig, bsig, out, N);
}